// GraphSAGE_65042984731050
// MI455X (gfx1250) — compile-verified
//
#include <hip/hip_runtime.h>
#include <hip/hip_bf16.h>

typedef __attribute__((ext_vector_type(2))) float v2f;
typedef __attribute__((ext_vector_type(8))) float v8f;

#define N_NODES 100000
#define N_EDGES 1600000

// ---------------------------------------------------------------------------
// Zero-fill kernel for workspace accumulators (d_ws is poisoned with 0xAA).
// ---------------------------------------------------------------------------
__global__ void zero_kernel(float* __restrict__ p, long long n) {
    long long i = (long long)blockIdx.x * blockDim.x + threadIdx.x;
    if (i < n) p[i] = 0.0f;
}

// ---------------------------------------------------------------------------
// Edge scatter: one thread per (edge, channel). Consecutive threads cover
// consecutive channels of the same edge -> coalesced gather of X[src] and a
// contiguous burst of L2 atomics at Agg[dst]. x/h/agg all fit in the 192MB L2,
// so the random edge indexing resolves on-chip rather than in HBM.
// ---------------------------------------------------------------------------
template<int C, bool COUNT_DEG>
__global__ void scatter_add_kernel(const float* __restrict__ X,
                                   const long long* __restrict__ src,
                                   const long long* __restrict__ dst,
                                   float* __restrict__ Agg,
                                   float* __restrict__ Deg,
                                   int nEdges) {
    long long idx = (long long)blockIdx.x * blockDim.x + threadIdx.x;
    long long e = idx / C;
    int c = (int)(idx % C);
    if (e >= nEdges) return;
    long long s = src[e];
    long long d = dst[e];
    atomicAdd(&Agg[(long long)d * C + c], X[(long long)s * C + c]);
    if (COUNT_DEG && c == 0) atomicAdd(&Deg[d], 1.0f);
}

// ---------------------------------------------------------------------------
// Fused SAGE linear: Out = act( (Agg * 1/max(deg,1)) @ Wl^T + X @ Wr^T + b )
// Full-precision fp32 WMMA (V_WMMA_F32_16X16X4_F32), wave32, one wave per
// 16-row strip of nodes, NT = NOUT/16 accumulator tiles held in registers.
//
// Fragment layouts per CDNA5 ISA 7.12.2 (32-bit):
//   A 16x4 : lane(0-15) row=m0+lane holds K={k0,k0+1}; lanes 16-31 hold
//            K={k0+2,k0+3} for the same rows.
//   B 4x16 : lane half selects K pair, (lane&15) selects column n. Since
//            B = W^T with W row-major [NOUT x K], each lane loads the two
//            adjacent-k floats W[col*K + ka], W[col*K + ka + 1].
//   C/D    : vgpr r -> row m0 + r + 8*(lane>>4), col = n0 + (lane&15).
// ---------------------------------------------------------------------------
template<int K, int NOUT, bool RELU>
__global__ __launch_bounds__(256) void sage_linear_wmma_kernel(
        const float* __restrict__ Agg,   // [M x K] un-normalized neighbor sums
        const float* __restrict__ Deg,   // [M]
        const float* __restrict__ X,     // [M x K] root features
        const float* __restrict__ Wl,    // [NOUT x K]
        const float* __restrict__ Wr,    // [NOUT x K]
        const float* __restrict__ Bias,  // [NOUT]
        float* __restrict__ Out,         // [M x NOUT]
        int M) {
    const int lane  = threadIdx.x & 31;
    const int wave  = threadIdx.x >> 5;
    const int strip = blockIdx.x * (blockDim.x >> 5) + wave;
    const int m0    = strip * 16;
    if (m0 >= M) return;                 // uniform per wave: EXEC stays all-1s

    const int  row   = m0 + (lane & 15);
    const int  khalf = (lane >> 4) * 2;  // 0 for lanes 0-15, 2 for lanes 16-31
    const int  ncol  = (lane & 15);
    const float sc   = 1.0f / fmaxf(Deg[row], 1.0f);   // mean normalization

    constexpr int NT = NOUT / 16;
    v8f acc[NT];
#pragma unroll
    for (int t = 0; t < NT; ++t)
#pragma unroll
        for (int i = 0; i < 8; ++i) acc[t][i] = 0.0f;

#pragma unroll
    for (int k0 = 0; k0 < K; k0 += 4) {
        const int ka = k0 + khalf;
        v2f a_l, a_r;
        a_l.x = Agg[(long long)row * K + ka]     * sc;
        a_l.y = Agg[(long long)row * K + ka + 1] * sc;
        a_r.x = X[(long long)row * K + ka];
        a_r.y = X[(long long)row * K + ka + 1];
#pragma unroll
        for (int t = 0; t < NT; ++t) {
            const int col = t * 16 + ncol;
            v2f b_l, b_r;
            b_l.x = Wl[col * K + ka];
            b_l.y = Wl[col * K + ka + 1];
            b_r.x = Wr[col * K + ka];
            b_r.y = Wr[col * K + ka + 1];
            acc[t] = __builtin_amdgcn_wmma_f32_16x16x4_f32(
                false, a_l, false, b_l, (short)0, acc[t], false, false);
            acc[t] = __builtin_amdgcn_wmma_f32_16x16x4_f32(
                false, a_r, false, b_r, (short)0, acc[t], false, false);
        }
    }

    const int rbase = m0 + ((lane >> 4) << 3);
#pragma unroll
    for (int t = 0; t < NT; ++t) {
        const int col = t * 16 + ncol;
        const float bv = Bias[col];
#pragma unroll
        for (int r = 0; r < 8; ++r) {
            float v = acc[t][r] + bv;
            if (RELU) v = fmaxf(v, 0.0f);
            Out[(long long)(rbase + r) * NOUT + col] = v;
        }
    }
}

// ---------------------------------------------------------------------------
// Launch
// ---------------------------------------------------------------------------
extern "C" void kernel_launch(void* const* d_in, const int* in_sizes, int n_in,
                              void* d_out, int out_size, void* d_ws, size_t ws_size,
                              hipStream_t stream) {
    const float*     x   = (const float*)d_in[0];
    const long long* ei  = (const long long*)d_in[1];  // [2 x N_EDGES] int64
    const float*     W1l = (const float*)d_in[2];
    const float*     W1r = (const float*)d_in[3];
    const float*     b1  = (const float*)d_in[4];
    const float*     W2l = (const float*)d_in[5];
    const float*     W2r = (const float*)d_in[6];
    const float*     b2  = (const float*)d_in[7];
    float*           out = (float*)d_out;

    const long long* srcI = ei;             // row 0
    const long long* dstI = ei + N_EDGES;   // row 1

    // Workspace layout (floats): [deg N][agg1 N*32][agg2 N*128][h N*128]
    float* ws   = (float*)d_ws;
    float* deg  = ws;
    float* agg1 = deg  + N_NODES;
    float* agg2 = agg1 + (long long)N_NODES * 32;
    float* h    = agg2 + (long long)N_NODES * 128;

    // 1) zero deg | agg1 | agg2 in one contiguous pass (h is fully overwritten)
    {
        long long zn = (long long)N_NODES * (1 + 32 + 128);
        long long blocks = (zn + 255) / 256;
        zero_kernel<<<(unsigned)blocks, 256, 0, stream>>>(ws, zn);
    }

    // 2) scatter layer 1 (32 ch) + degree count
    {
        long long total = (long long)N_EDGES * 32;
        long long blocks = (total + 255) / 256;
        scatter_add_kernel<32, true><<<(unsigned)blocks, 256, 0, stream>>>(
            x, srcI, dstI, agg1, deg, N_EDGES);
    }

    // 3) h = relu(agg1/deg @ W1l^T + x @ W1r^T + b1)   [K=32, NOUT=128]
    {
        int strips = (N_NODES + 15) / 16;          // 6250
        int blocks = (strips + 7) / 8;             // 8 waves / block
        sage_linear_wmma_kernel<32, 128, true><<<blocks, 256, 0, stream>>>(
            agg1, deg, x, W1l, W1r, b1, h, N_NODES);
    }

    // 4) scatter layer 2 (128 ch), degree already computed
    {
        long long total = (long long)N_EDGES * 128;
        long long blocks = (total + 255) / 256;
        scatter_add_kernel<128, false><<<(unsigned)blocks, 256, 0, stream>>>(
            h, srcI, dstI, agg2, deg, N_EDGES);
    }

    // 5) out = agg2/deg @ W2l^T + h @ W2r^T + b2       [K=128, NOUT=32]
    {
        int strips = (N_NODES + 15) / 16;
        int blocks = (strips + 7) / 8;
        sage_linear_wmma_kernel<128, 32, false><<<blocks, 256, 0, stream>>>(
            agg2, deg, h, W2l, W2r, b2, out, N_NODES);
    }
}